// GCN_55456617726008
// MI455X (gfx1250) — compile-verified
//
#include <hip/hip_runtime.h>
#include <hip/hip_bf16.h>

#define HDIM   128
#define NNODES 50000
#define NEDGES 600000
#define NGRAPH 256
#define BN_EPS 1e-5f

#define MT         5            // M-tiles (16 rows each) per wave -> 80 rows per block
#define MROWS      (MT * 16)    // 80
#define LDS_STRIDE 132          // 128 + 4 pad floats: stride-4 banks mod 64 -> conflict-free

typedef float v2f __attribute__((ext_vector_type(2)));
typedef float v8f __attribute__((ext_vector_type(8)));

// ---------------- utility ----------------
__global__ void zero_f32(float* __restrict__ p, int n) {
  int i = blockIdx.x * blockDim.x + threadIdx.x;
  if (i < n) p[i] = 0.0f;
}

// ---------------- degree / normalization ----------------
__global__ void deg_kernel(const int* __restrict__ dst, float* __restrict__ deg, int E) {
  int e = blockIdx.x * blockDim.x + threadIdx.x;
  if (e < E) atomicAdd(&deg[dst[e]], 1.0f);
}

__global__ void dis_kernel(float* __restrict__ deg, int n) {
  int i = blockIdx.x * blockDim.x + threadIdx.x;
  if (i < n) deg[i] = rsqrtf(deg[i] + 1.0f);
}

// ---------------- fp32 WMMA GEMM: Y[NNODES,128] = X[NNODES,128] * W[128,128] ----------------
// Block = 256 threads (8 waves). Block owns an 80-row M strip; wave w owns N-tile n0=w*16
// and computes MT=5 stacked 16x16 tiles, so each B fragment load feeds 5 WMMAs.
// The X strip (80x128 f32 = 40KB) is staged to LDS once per block with CDNA5 async
// global->LDS loads (ASYNCcnt), then A fragments come from ds_load_b64 (padded stride).
__global__ void __launch_bounds__(256) gemm_wmma(const float* __restrict__ X,
                                                 const float* __restrict__ W,
                                                 float* __restrict__ Y) {
  __shared__ float lds_x[MROWS * LDS_STRIDE];   // 42240 B
  const int tid = threadIdx.x;
  const int m0  = blockIdx.x * MROWS;           // 625 blocks * 80 = 50000 exactly

  // ---- stage X strip: 80 rows x 512B = 2560 x 16B chunks, 10 per thread ----
  const float* xbase = X + (size_t)m0 * HDIM;
  const unsigned lds0 = (unsigned)(uintptr_t)(&lds_x[0]);   // as3 offset = low 32 bits
#pragma unroll
  for (int j = 0; j < 10; ++j) {
    int c     = tid + j * 256;
    int row   = c >> 5;             // 32 chunks per row
    int inrow = (c & 31) * 16;      // byte offset within row
    int goff  = row * (HDIM * 4) + inrow;
    int loff  = (int)lds0 + row * (LDS_STRIDE * 4) + inrow;
    asm volatile("global_load_async_to_lds_b128 %0, %1, %2"
                 :: "v"(loff), "v"(goff), "s"(xbase) : "memory");
  }
  asm volatile("s_wait_asynccnt 0" ::: "memory");
  __syncthreads();

  // ---- compute ----
  const int wave = tid >> 5;
  const int lane = tid & 31;
  const int half = lane >> 4;       // 0/1: selects K pair {k,k+1} vs {k+2,k+3}
  const int l15  = lane & 15;
  const int n0   = wave * 16;
  const int col  = n0 + l15;

  v8f acc[MT];
#pragma unroll
  for (int t = 0; t < MT; ++t) acc[t] = v8f{0.f, 0.f, 0.f, 0.f, 0.f, 0.f, 0.f, 0.f};

#pragma unroll
  for (int k = 0; k < HDIM; k += 4) {
    const int ka = k + 2 * half;
    v2f b;
    b.x = W[ka * HDIM + col];
    b.y = W[(ka + 1) * HDIM + col];
#pragma unroll
    for (int t = 0; t < MT; ++t) {
      const float* ar = &lds_x[(t * 16 + l15) * LDS_STRIDE + ka];
      v2f a;
      a.x = ar[0];
      a.y = ar[1];
      acc[t] = __builtin_amdgcn_wmma_f32_16x16x4_f32(false, a, false, b,
                                                     (short)0, acc[t], false, false);
    }
  }

  const int rbase0 = m0 + half * 8;
#pragma unroll
  for (int t = 0; t < MT; ++t) {
    const int rbase = rbase0 + t * 16;
#pragma unroll
    for (int r = 0; r < 8; ++r)
      Y[(size_t)(rbase + r) * HDIM + col] = acc[t][r];
  }
}

// ---------------- out = dis^2 * h + bias (self-loop term, initializes accumulator) ----------------
__global__ void self_bias(const float* __restrict__ h, const float* __restrict__ dis,
                          const float* __restrict__ bias, float* __restrict__ out, int n) {
  int idx = blockIdx.x * blockDim.x + threadIdx.x;   // over n * (HDIM/4) float4s
  int total = n * (HDIM / 4);
  if (idx >= total) return;
  int row = idx >> 5;          // HDIM/4 == 32
  int c4  = idx & 31;
  float d = dis[row];
  float w = d * d;
  float4 v = ((const float4*)h)[idx];
  float4 b = ((const float4*)bias)[c4];
  float4 o;
  o.x = w * v.x + b.x;
  o.y = w * v.y + b.y;
  o.z = w * v.z + b.z;
  o.w = w * v.w + b.w;
  ((float4*)out)[idx] = o;
}

// ---------------- edge scatter: out[dst] += dis[src]*dis[dst]*h[src] ----------------
__global__ void __launch_bounds__(256) edge_scatter(const float* __restrict__ h,
                                                    const float* __restrict__ dis,
                                                    const int* __restrict__ src,
                                                    const int* __restrict__ dst,
                                                    float* __restrict__ out, int E) {
  int e = blockIdx.x * (blockDim.x >> 5) + (threadIdx.x >> 5);
  if (e >= E) return;
  int lane = threadIdx.x & 31;
  int s = src[e];
  int d = dst[e];
  float w = dis[s] * dis[d];
  float4 v = ((const float4*)(h + (size_t)s * HDIM))[lane];
  float* o = out + (size_t)d * HDIM + lane * 4;
  atomicAdd(o + 0, w * v.x);
  atomicAdd(o + 1, w * v.y);
  atomicAdd(o + 2, w * v.z);
  atomicAdd(o + 3, w * v.w);
}

// ---------------- batchnorm stats: per-channel sum / sumsq ----------------
__global__ void __launch_bounds__(128) bn_stats(const float* __restrict__ x,
                                                float* __restrict__ sums,
                                                float* __restrict__ sumsq,
                                                int n, int rows_per_block) {
  int c  = threadIdx.x;
  int r0 = blockIdx.x * rows_per_block;
  int r1 = r0 + rows_per_block;
  if (r1 > n) r1 = n;
  float s = 0.f, q = 0.f;
  for (int r = r0; r < r1; ++r) {
    float v = x[(size_t)r * HDIM + c];
    s += v;
    q += v * v;
  }
  atomicAdd(&sums[c], s);
  atomicAdd(&sumsq[c], q);
}

__global__ void bn_apply_relu(float* __restrict__ x, const float* __restrict__ sums,
                              const float* __restrict__ sumsq, const float* __restrict__ g,
                              const float* __restrict__ be, int n) {
  int idx = blockIdx.x * blockDim.x + threadIdx.x;
  if (idx >= n * HDIM) return;
  int c = idx & (HDIM - 1);
  float invn = 1.0f / (float)n;
  float m = sums[c] * invn;
  float v = sumsq[c] * invn - m * m;
  float y = g[c] * (x[idx] - m) * rsqrtf(v + BN_EPS) + be[c];
  x[idx] = fmaxf(y, 0.0f);
}

__global__ void relu_k(float* __restrict__ x, int n) {
  int i = blockIdx.x * blockDim.x + threadIdx.x;
  if (i < n) x[i] = fmaxf(x[i], 0.0f);
}

// ---------------- global mean pool ----------------
__global__ void __launch_bounds__(256) pool_scatter(const float* __restrict__ h,
                                                    const int* __restrict__ batch,
                                                    float* __restrict__ pool,
                                                    float* __restrict__ cnt, int n) {
  int node = blockIdx.x * (blockDim.x >> 5) + (threadIdx.x >> 5);
  if (node >= n) return;
  int lane = threadIdx.x & 31;
  int g = batch[node];
  float4 v = ((const float4*)(h + (size_t)node * HDIM))[lane];
  float* p = pool + (size_t)g * HDIM + lane * 4;
  atomicAdd(p + 0, v.x);
  atomicAdd(p + 1, v.y);
  atomicAdd(p + 2, v.z);
  atomicAdd(p + 3, v.w);
  if (lane == 0) atomicAdd(&cnt[g], 1.0f);
}

__global__ void final_linear(const float* __restrict__ pool, const float* __restrict__ cnt,
                             const float* __restrict__ Wl, const float* __restrict__ bl,
                             float* __restrict__ out) {
  int g = blockIdx.x * blockDim.x + threadIdx.x;
  if (g >= NGRAPH) return;
  float s = 0.f;
  for (int k = 0; k < HDIM; ++k) s += pool[(size_t)g * HDIM + k] * Wl[k];
  out[g] = s / fmaxf(cnt[g], 1.0f) + bl[0];
}

// ---------------- host side ----------------
static inline size_t alignUp(size_t v) { return (v + 255) & ~(size_t)255; }

extern "C" void kernel_launch(void* const* d_in, const int* in_sizes, int n_in,
                              void* d_out, int out_size, void* d_ws, size_t ws_size,
                              hipStream_t stream) {
  (void)in_sizes; (void)n_in; (void)out_size; (void)ws_size;
  const float* x     = (const float*)d_in[0];
  const int*   ei    = (const int*)d_in[1];
  const int*   src   = ei;              // edge_index row 0
  const int*   dst   = ei + NEDGES;     // edge_index row 1
  const int*   batch = (const int*)d_in[2];
  const float* W1  = (const float*)d_in[3];
  const float* b1  = (const float*)d_in[4];
  const float* g1  = (const float*)d_in[5];
  const float* be1 = (const float*)d_in[6];
  const float* W2  = (const float*)d_in[7];
  const float* b2  = (const float*)d_in[8];
  const float* g2  = (const float*)d_in[9];
  const float* be2 = (const float*)d_in[10];
  const float* W3  = (const float*)d_in[11];
  const float* b3  = (const float*)d_in[12];
  const float* Wl  = (const float*)d_in[13];
  const float* bl  = (const float*)d_in[14];
  float* out = (float*)d_out;

  // workspace carve-up
  char* ws = (char*)d_ws;
  const size_t nh = (size_t)NNODES * HDIM;
  float* bufA  = (float*)ws;  ws += alignUp(nh * sizeof(float));   // gemm output h
  float* bufB  = (float*)ws;  ws += alignUp(nh * sizeof(float));   // aggregation output
  float* dis   = (float*)ws;  ws += alignUp((size_t)NNODES * sizeof(float));
  float* stats = (float*)ws;  ws += alignUp(2 * HDIM * sizeof(float)); // sums | sumsq
  float* pool  = (float*)ws;  ws += alignUp((size_t)NGRAPH * HDIM * sizeof(float));
  float* cnt   = (float*)ws;  ws += alignUp((size_t)NGRAPH * sizeof(float));
  float* sums  = stats;
  float* sumsq = stats + HDIM;

  const int NH = NNODES * HDIM;                 // 6,400,000
  const int gridElems = (NH + 255) / 256;
  const int gridSelf  = (NNODES * (HDIM / 4) + 255) / 256;
  const int gridEdgeW = (NEDGES + 7) / 8;       // 8 edge-waves per 256-thread block
  const int gridNodeW = (NNODES + 7) / 8;
  const int rowsPerBlk = (NNODES + 255) / 256;

  // --- degrees & symmetric normalization ---
  zero_f32<<<(NNODES + 255) / 256, 256, 0, stream>>>(dis, NNODES);
  deg_kernel<<<(NEDGES + 255) / 256, 256, 0, stream>>>(dst, dis, NEDGES);
  dis_kernel<<<(NNODES + 255) / 256, 256, 0, stream>>>(dis, NNODES);

  auto layer = [&](const float* in, const float* Wt, const float* bt,
                   const float* gt, const float* bet, bool do_bn) {
    gemm_wmma<<<NNODES / MROWS, 256, 0, stream>>>(in, Wt, bufA);
    self_bias<<<gridSelf, 256, 0, stream>>>(bufA, dis, bt, bufB, NNODES);
    edge_scatter<<<gridEdgeW, 256, 0, stream>>>(bufA, dis, src, dst, bufB, NEDGES);
    if (do_bn) {
      zero_f32<<<1, 256, 0, stream>>>(stats, 2 * HDIM);
      bn_stats<<<256, 128, 0, stream>>>(bufB, sums, sumsq, NNODES, rowsPerBlk);
      bn_apply_relu<<<gridElems, 256, 0, stream>>>(bufB, sums, sumsq, gt, bet, NNODES);
    } else {
      relu_k<<<gridElems, 256, 0, stream>>>(bufB, NH);
    }
  };

  layer(x,    W1, b1, g1, be1, true);
  layer(bufB, W2, b2, g2, be2, true);
  layer(bufB, W3, b3, nullptr, nullptr, false);

  // --- global mean pool + linear head ---
  zero_f32<<<(NGRAPH * HDIM + 255) / 256, 256, 0, stream>>>(pool, NGRAPH * HDIM);
  zero_f32<<<1, 256, 0, stream>>>(cnt, NGRAPH);
  pool_scatter<<<gridNodeW, 256, 0, stream>>>(bufB, batch, pool, cnt, NNODES);
  final_linear<<<1, 256, 0, stream>>>(pool, cnt, Wl, bl, out);
}